// Toeplitz_68667937129004
// MI455X (gfx1250) — compile-verified
//
#include <hip/hip_runtime.h>
#include <hip/hip_bf16.h>

// Problem constants (reference: B=2, L=2048, 16 heads x 128 dim)
#define LSEQ    2048
#define DMODEL  2048
#define NHEADSC 16
#define HDIM    128
#define KLEN    4095            // 2*LSEQ - 1
#define MT      128             // output rows per workgroup
#define KC      32              // t-chunk staged in LDS per iteration
#define NCHUNK  (LSEQ / KC)     // 64
#define THREADS 256             // 8 wave32s

typedef float v2f __attribute__((ext_vector_type(2)));
typedef float v4f __attribute__((ext_vector_type(4)));
typedef float v8f __attribute__((ext_vector_type(8)));

// O[m, h] = sum_t T[m,t] * X[t, h]  with  T[m,t] = conv[(m - t + LSEQ) mod KLEN]
// per (batch, head); then out = std_dev * O + residual(x).
//
// LDS X layout is K-pair interleaved:  addr(k, col) = ((k>>1)*128 + col)*2 + (k&1)
// -> each WMMA B fragment {X[k][col], X[k+1][col]} is one ds_load_b64 into an
//    even VGPR pair (bank-conflict-free: upper lane half lands +32 banks).
// Conv generator is staged REVERSED: s_cr[j] = conv[KLEN - j] (j=0 is the mod-K
// wrap element) -> the A fragment {conv[idx], conv[idx-1]} becomes the ascending
// adjacent pair {s_cr[jdx], s_cr[jdx+1]} (single ds_load_2addr, no swap movs),
// and jdx grows with the k-step so per-step deltas fold into DS immediate offsets.
__launch_bounds__(THREADS)
__global__ void toeplitz_wmma_f32(const float* __restrict__ x,
                                  const float* __restrict__ fconv,
                                  const float* __restrict__ rconv,
                                  float* __restrict__ out)
{
    const int mt   = blockIdx.x;            // m-tile: 0..15
    const int head = blockIdx.y;            // 0..15
    const int b    = blockIdx.z;            // batch
    const int tid  = (int)threadIdx.x;
    const int lane = tid & 31;
    const int wave = tid >> 5;
    const int nl   = lane & 15;             // N / M position within lane half
    const int half = lane >> 4;             // 0 or 1 (lane half)
    const int kb   = half << 1;             // K offset of this lane half: 0 or 2

    __shared__ float s_cr[4096];            // reversed conv: s_cr[j] = conv[(KLEN - j) mod KLEN]
    __shared__ float s_x[2][KC * HDIM];     // double-buffered, K-pair interleaved

    // ---- stage reversed Toeplitz generator vector into LDS ----
    const size_t cb = (size_t)b * LSEQ * NHEADSC + head;
    #pragma unroll
    for (int i = 0; i < 16; ++i) {
        int j = tid + i * THREADS;          // 0..4095
        float v;
        if (j > LSEQ)       v = rconv[cb + (size_t)(j - LSEQ) * NHEADSC];        // rc[1..2047]
        else if (j >= 1)    v = fconv[cb + (size_t)(LSEQ - j) * NHEADSC];        // fc[0..2047]
        else                v = rconv[cb + (size_t)(LSEQ - 1) * NHEADSC];        // wrap: conv[0]=rc[L-1]
        s_cr[j] = v;
    }

    // X panel for this (b, head): row t -> xb + t*DMODEL, 128 contiguous floats
    const float* xb = x + (size_t)b * LSEQ * DMODEL + (size_t)head * HDIM;

    // chunk staging: 16 row-pairs x 32 col4-groups = 512 cells; 2 cells/thread.
    // cell: load rows 2p, 2p+1 (v4f each), interleave, 2x ds_store_b128.
    const int p0  = tid >> 5;               // first row-pair  (== wave)
    const int p1  = p0 + 8;                 // second row-pair
    const int cc4 = (tid & 31) << 2;        // starting column of this thread's 4 cols
    v4f ra[2], rb[2];

    auto load_chunk = [&](int t0) {
        const float* g0 = xb + (size_t)(t0 + 2 * p0) * DMODEL + cc4;
        const float* g1 = xb + (size_t)(t0 + 2 * p1) * DMODEL + cc4;
        ra[0] = *(const v4f*)g0;  rb[0] = *(const v4f*)(g0 + DMODEL);
        ra[1] = *(const v4f*)g1;  rb[1] = *(const v4f*)(g1 + DMODEL);
    };
    auto store_chunk = [&](int buf) {
        #pragma unroll
        for (int i = 0; i < 2; ++i) {
            v4f lo, hi;
            lo.x = ra[i].x; lo.y = rb[i].x; lo.z = ra[i].y; lo.w = rb[i].y;
            hi.x = ra[i].z; hi.y = rb[i].z; hi.z = ra[i].w; hi.w = rb[i].w;
            float* d = &s_x[buf][((i ? p1 : p0) * HDIM + cc4) << 1];
            *(v4f*)d       = lo;
            *(v4f*)(d + 4) = hi;
        }
    };

    load_chunk(0);
    store_chunk(0);
    __syncthreads();

    v8f acc[8];
    {
        v8f z = {0.f,0.f,0.f,0.f,0.f,0.f,0.f,0.f};
        #pragma unroll
        for (int i = 0; i < 8; ++i) acc[i] = z;
    }

    const int mrow  = mt * MT + wave * 16 + nl;   // A fragment row (M = nl within lane half)
    // conv index: idx = (mrow + LSEQ - kb) - t_abs; reversed: jdx = KLEN - idx
    const int jbase = (LSEQ - 1) - mrow + kb;     // jdx = jbase + t0 + 4s

    for (int c = 0; c < NCHUNK; ++c) {
        const int buf = c & 1;
        const int t0  = c * KC;
        if (c + 1 < NCHUNK) load_chunk(t0 + KC);  // prefetch next chunk to registers

        const v2f* sx2 = (const v2f*)&s_x[buf][0];
        const float* cr = &s_cr[jbase + t0];      // one base per chunk; per-step offsets immediate
        #pragma unroll
        for (int s = 0; s < KC / 4; ++s) {
            // A: {conv[idx], conv[idx-1]} == ascending pair {cr[4s], cr[4s+1]}
            v2f A;
            A.x = cr[(s << 2)];
            A.y = cr[(s << 2) + 1];
            // B base: pair-row (2s + half), column nl (+ nt*16 as immediate offset)
            const v2f* bp = sx2 + ((2 * s + half) * HDIM + nl);
            #pragma unroll
            for (int nt = 0; nt < 8; ++nt) {
                v2f Bv = bp[nt * 16];            // one ds_load_b64 -> even VGPR pair
                acc[nt] = __builtin_amdgcn_wmma_f32_16x16x4_f32(
                    false, A, false, Bv, (short)0, acc[nt], false, false);
            }
        }
        if (c + 1 < NCHUNK) store_chunk(buf ^ 1); // fill the other buffer
        __syncthreads();
    }

    // ---- epilogue: out = std_dev * acc + residual ----
    const float sd = 0.011048543456039806f;       // 0.5 / sqrt(2048)
    const int mb = mt * MT + wave * 16 + (half << 3);  // C layout: upper lanes are M+8
    #pragma unroll
    for (int nt = 0; nt < 8; ++nt) {
        const size_t col = (size_t)head * HDIM + nt * 16 + nl;
        #pragma unroll
        for (int r = 0; r < 8; ++r) {
            const size_t o = ((size_t)b * LSEQ + (size_t)(mb + r)) * DMODEL + col;
            out[o] = sd * acc[nt][r] + x[o];
        }
    }
}

extern "C" void kernel_launch(void* const* d_in, const int* in_sizes, int n_in,
                              void* d_out, int out_size, void* d_ws, size_t ws_size,
                              hipStream_t stream) {
    const float* x  = (const float*)d_in[0];   // (B, L, 2048) f32
    const float* fc = (const float*)d_in[1];   // (B, L, 16)   f32
    const float* rc = (const float*)d_in[2];   // (B, L, 16)   f32
    float* out = (float*)d_out;                // (B, L, 2048) f32
    const int B = in_sizes[0] / (LSEQ * DMODEL);
    dim3 grid(LSEQ / MT, NHEADSC, B);
    toeplitz_wmma_f32<<<grid, THREADS, 0, stream>>>(x, fc, rc, out);
}